// Graph_Cluster_Module_53446573031444
// MI455X (gfx1250) — compile-verified
//
#include <hip/hip_runtime.h>

// Problem constants (match reference)
#define HH 1024
#define WW 1536
#define NN (HH * WW)
#define SIZE_THRC 256
#define CC_ITERS 300

// LDS-tiled multi-iteration dilation
#define TILE 32
#define HALO 4
#define ITER_PER_LAUNCH 4
#define REG (TILE + 2 * HALO)      // 40
#define NLAUNCH (CC_ITERS / ITER_PER_LAUNCH)  // 75

typedef __attribute__((ext_vector_type(2))) float v2f;
typedef __attribute__((ext_vector_type(8))) float v8f;

__device__ __forceinline__ int imin(int a, int b) { return a < b ? a : b; }
__device__ __forceinline__ int imax(int a, int b) { return a > b ? a : b; }

// ---------------------------------------------------------------- zero
__global__ void k_zero(int* __restrict__ p, int n) {
    int i = blockIdx.x * blockDim.x + threadIdx.x;
    if (i < n) p[i] = 0;
}

// ------------------------------------------------- fore mask + dst ids
__global__ void k_prep(const float* __restrict__ y_hat, const float* __restrict__ df,
                       int* __restrict__ dst, int* __restrict__ fore,
                       int* __restrict__ y1, int* __restrict__ y2) {
    int i = blockIdx.x * blockDim.x + threadIdx.x;
    if (i >= NN) return;
    int r = i / WW;
    int c = i - r * WW;
    float d0 = df[i];        // row displacement
    float d1 = df[NN + i];   // col displacement
    float len = sqrtf(__fadd_rn(__fmul_rn(d0, d0), __fmul_rn(d1, d1)));
    float den = __fadd_rn(len, 1e-19f);
    float u0 = __fdiv_rn(d0, den);
    float u1 = __fdiv_rn(d1, den);
    float dfc = fminf(fmaxf(len, 3.0f), 6.0f);
    float rr = __fadd_rn((float)r, __fmul_rn(u0, dfc));
    float cc = __fadd_rn((float)c, __fmul_rn(u1, dfc));
    rr = fminf(fmaxf(rr, 0.0f), (float)(HH - 1));
    cc = fminf(fmaxf(cc, 0.0f), (float)(WW - 1));
    int ir = (int)rintf(rr);            // round-half-even == jnp.round
    int ic = (int)rintf(cc);
    dst[i]  = ir * WW + ic;
    fore[i] = (y_hat[i] > 0.5f) ? 1 : 0;
    y1[i] = 0;
    y2[i] = 0;
}

// ------------------------------------------------- segment_sum (exact int)
__global__ void k_scatter1(const int* __restrict__ fore, const int* __restrict__ dst,
                           int* __restrict__ y1) {
    int i = blockIdx.x * blockDim.x + threadIdx.x;
    if (i >= NN) return;
    if (fore[i]) atomicAdd(&y1[dst[i]], 1);
}

__global__ void k_scatter2(const int* __restrict__ y1, const int* __restrict__ dst,
                           int* __restrict__ y2) {
    int i = blockIdx.x * blockDim.x + threadIdx.x;
    if (i >= NN) return;
    int v = y1[i];
    if (v) atomicAdd(&y2[dst[i]], v);
}

// ------------------------------------------------- horizontal 5-sum
__global__ void k_hsum(const int* __restrict__ y2, float* __restrict__ srow) {
    int i = blockIdx.x * blockDim.x + threadIdx.x;
    if (i >= NN) return;
    int r = i / WW;
    int c = i - r * WW;
    int c0 = imax(c - 2, 0), c1 = imin(c + 2, WW - 1);
    int s = 0;
    const int* row = y2 + r * WW;
    for (int cc = c0; cc <= c1; ++cc) s += row[cc];
    srow[i] = (float)s;   // exact: s < 2^24
}

// ---------------------------------------------------------------- WMMA pool
// Vertical 5-sum as banded matmul per 16x16 tile:
//   D(16x16) = sum_{q=0..4} A_q(16x4) x B_q(4x16)
// A[m][kg] = 1 iff kg in [m, m+4], kg = input-row - (r0-2), kg in 0..19.
// Then threshold: avgpool(y) >= 0.5  <=>  2*s >= cv*ch (count_include_pad=False).
__global__ void k_pool_wmma(const float* __restrict__ srow, int* __restrict__ mask0) {
    int lane = threadIdx.x & 31;
    int wave = (blockIdx.x * blockDim.x + threadIdx.x) >> 5;
    const int TX = WW / 16;                 // 96 tiles per row
    int tr = wave / TX;
    int tc = wave - tr * TX;
    if (tr >= HH / 16) return;              // wave-uniform; never taken (grid exact)
    int r0 = tr * 16, c0 = tc * 16;
    int m    = lane & 15;                   // A: M index / B: N index
    int half = lane >> 4;                   // lane-half splits K range

    v8f acc = {};
#if __has_builtin(__builtin_amdgcn_wmma_f32_16x16x4_f32)
#pragma unroll
    for (int q = 0; q < 5; ++q) {
        int kg0 = 4 * q + 2 * half;
        v2f a, b;
        a.x = (kg0     >= m && kg0     <= m + 4) ? 1.0f : 0.0f;
        a.y = (kg0 + 1 >= m && kg0 + 1 <= m + 4) ? 1.0f : 0.0f;
        int row0 = r0 - 2 + kg0;
        int col  = c0 + m;
        b.x = (row0     >= 0 && row0     < HH) ? srow[row0 * WW + col]       : 0.0f;
        b.y = (row0 + 1 >= 0 && row0 + 1 < HH) ? srow[(row0 + 1) * WW + col] : 0.0f;
        acc = __builtin_amdgcn_wmma_f32_16x16x4_f32(
            /*neg_a=*/false, a, /*neg_b=*/false, b,
            /*c_mod=*/(short)0, acc, /*reuse_a=*/false, /*reuse_b=*/false);
    }
#else
#pragma unroll
    for (int q = 0; q < 5; ++q) {
        int kg0 = 4 * q + 2 * half;
        for (int kk = 0; kk < 2; ++kk) {
            int kg = kg0 + kk;
            int row = r0 - 2 + kg;
            float v = (row >= 0 && row < HH) ? srow[row * WW + c0 + m] : 0.0f;
            for (int j = 0; j < 8; ++j) {
                int mo = j + 8 * half;
                if (kg >= mo && kg <= mo + 4) acc[j] += v;
            }
        }
    }
#endif

#pragma unroll
    for (int j = 0; j < 8; ++j) {
        int mo = j + 8 * half;              // D layout: VGPR j -> M = j (+8 for hi lanes)
        int r = r0 + mo, c = c0 + m;
        int cv = imin(r + 2, HH - 1) - imax(r - 2, 0) + 1;
        int ch = imin(c + 2, WW - 1) - imax(c - 2, 0) + 1;
        float s = acc[j];
        mask0[r * WW + c] = (2.0f * s >= (float)(cv * ch)) ? 1 : 0;
    }
}

// ------------------------------------------------- erode (cross, border=fg) + label init
__global__ void k_erode_label(const int* __restrict__ m0, int* __restrict__ lab) {
    int i = blockIdx.x * blockDim.x + threadIdx.x;
    if (i >= NN) return;
    int r = i / WW;
    int c = i - r * WW;
    int v = m0[i];
    if (v) {
        int up = (r > 0)      ? m0[i - WW] : 1;
        int dn = (r < HH - 1) ? m0[i + WW] : 1;
        int lf = (c > 0)      ? m0[i - 1]  : 1;
        int rt = (c < WW - 1) ? m0[i + 1]  : 1;
        v = v & up & dn & lf & rt;
    }
    lab[i] = v ? (i + 1) : 0;
}

// ------------------------------------------------- LDS-tiled 4x max-dilation
// Each block: load (TILE+2*HALO)^2 region into LDS (zero-padded at image
// borders == reduce_window's 0-padding), run ITER_PER_LAUNCH local dilation
// steps (ghost-zone: after t steps, cells >= t from LDS border are exact,
// HALO = ITER_PER_LAUNCH), write the exact TILE x TILE interior.
// Invariant: lab > 0  <=>  eroded mask, so the mask is implicit.
__global__ void k_dilate4(const int* __restrict__ in, int* __restrict__ out) {
    __shared__ int t0[REG * REG];
    __shared__ int t1[REG * REG];
    const int bx = blockIdx.x, by = blockIdx.y;
    const int gx0 = bx * TILE - HALO;
    const int gy0 = by * TILE - HALO;
    const int tid = threadIdx.x;
    const int nthr = blockDim.x;

    for (int idx = tid; idx < REG * REG; idx += nthr) {
        int ly = idx / REG, lx = idx - ly * REG;
        int gy = gy0 + ly, gx = gx0 + lx;
        int v = 0;
        if (gy >= 0 && gy < HH && gx >= 0 && gx < WW) v = in[gy * WW + gx];
        t0[idx] = v;
    }
    __syncthreads();

    int* src = t0;
    int* dstb = t1;
#pragma unroll
    for (int it = 0; it < ITER_PER_LAUNCH; ++it) {
        for (int idx = tid; idx < REG * REG; idx += nthr) {
            int ly = idx / REG, lx = idx - ly * REG;
            int v = src[idx];
            int o;
            if (ly == 0 || ly == REG - 1 || lx == 0 || lx == REG - 1) {
                o = v;                       // border copy-through (stale ok)
            } else if (v == 0) {
                o = 0;                       // not in mask
            } else {
                int mx = v;
                mx = imax(mx, src[idx - REG - 1]);
                mx = imax(mx, src[idx - REG]);
                mx = imax(mx, src[idx - REG + 1]);
                mx = imax(mx, src[idx - 1]);
                mx = imax(mx, src[idx + 1]);
                mx = imax(mx, src[idx + REG - 1]);
                mx = imax(mx, src[idx + REG]);
                mx = imax(mx, src[idx + REG + 1]);
                o = mx;
            }
            dstb[idx] = o;
        }
        __syncthreads();
        int* t = src; src = dstb; dstb = t;
    }
    // ITER_PER_LAUNCH is even -> result back in t0 (== src)

    for (int idx = tid; idx < TILE * TILE; idx += nthr) {
        int ly = idx / TILE, lx = idx - ly * TILE;
        int gy = by * TILE + ly, gx = bx * TILE + lx;
        out[gy * WW + gx] = src[(ly + HALO) * REG + (lx + HALO)];
    }
}

// ------------------------------------------------- pointer jumping: g = f . f
__global__ void k_compose(const int* __restrict__ f, int* __restrict__ g) {
    int i = blockIdx.x * blockDim.x + threadIdx.x;
    if (i >= NN) return;
    g[i] = f[f[i]];
}

// ------------------------------------------------- per-label component counts
__global__ void k_count(const int* __restrict__ fore, const int* __restrict__ lab,
                        const int* __restrict__ dst8, int* __restrict__ li,
                        int* __restrict__ counts) {
    int i = blockIdx.x * blockDim.x + threadIdx.x;
    if (i >= NN) return;
    int l = fore[i] ? lab[dst8[i]] : 0;
    li[i] = l;
    if (l > 0) atomicAdd(&counts[l], 1);
}

// ------------------------------------------------- size-threshold broadcast
__global__ void k_final(const int* __restrict__ li, const int* __restrict__ counts,
                        float* __restrict__ out) {
    int i = blockIdx.x * blockDim.x + threadIdx.x;
    if (i >= NN) return;
    int l = li[i];
    out[i] = (l > 0 && counts[l] > SIZE_THRC) ? (float)l : 0.0f;
}

extern "C" void kernel_launch(void* const* d_in, const int* in_sizes, int n_in,
                              void* d_out, int out_size, void* d_ws, size_t ws_size,
                              hipStream_t stream) {
    (void)in_sizes; (void)n_in; (void)out_size; (void)ws_size;
    const float* y_hat = (const float*)d_in[0];   // (H, W) float32
    const float* df    = (const float*)d_in[1];   // (2, H, W) float32
    float* out = (float*)d_out;

    char* ws = (char*)d_ws;
    size_t off = 0;
    auto salloc = [&](size_t bytes) -> void* {
        void* p = (void*)(ws + off);
        off += (bytes + 255) & ~(size_t)255;
        return p;
    };
    int*   dst    = (int*)salloc(sizeof(int) * NN);
    int*   fore   = (int*)salloc(sizeof(int) * NN);
    int*   bufA   = (int*)salloc(sizeof(int) * NN);        // y1 -> dst^2 -> dst^8
    int*   bufB   = (int*)salloc(sizeof(int) * NN);        // y2 -> dst^4
    float* srow   = (float*)salloc(sizeof(float) * NN);    // hsum -> li (int reuse)
    int*   mask0  = (int*)salloc(sizeof(int) * NN);
    int*   labA   = (int*)salloc(sizeof(int) * NN);
    int*   labB   = (int*)salloc(sizeof(int) * NN);
    int*   counts = (int*)salloc(sizeof(int) * (NN + 64));

    dim3 B(256);
    int GB = (NN + 255) / 256;

    k_prep    <<<GB, B, 0, stream>>>(y_hat, df, dst, fore, bufA, bufB);
    k_scatter1<<<GB, B, 0, stream>>>(fore, dst, bufA);
    k_scatter2<<<GB, B, 0, stream>>>(bufA, dst, bufB);
    k_hsum    <<<GB, B, 0, stream>>>(bufB, srow);

    // 6144 tiles, one wave (32 lanes) each; 8 waves/block -> 768 blocks, exact.
    k_pool_wmma<<<(HH / 16) * (WW / 16) / 8, 256, 0, stream>>>(srow, mask0);

    k_erode_label<<<GB, B, 0, stream>>>(mask0, labA);

    // NLAUNCH x ITER_PER_LAUNCH = CC_ITERS dilation steps, bit-identical to the
    // reference's capped while-loop (post-fixpoint dilations are identity).
    // Ghost-zone LDS tiling: 4 global-equivalent steps per launch.
    dim3 tgrid(WW / TILE, HH / TILE);   // 48 x 32
    int* pin = labA;
    int* pout = labB;
    for (int it = 0; it < NLAUNCH; ++it) {
        k_dilate4<<<tgrid, B, 0, stream>>>(pin, pout);
        int* t = pin; pin = pout; pout = t;
    }
    // final labels are in `pin` (NLAUNCH = 75 -> labB)

    // dst^8 via pointer jumping (3 gathers instead of 8)
    k_compose<<<GB, B, 0, stream>>>(dst, bufA);    // bufA = dst^2
    k_compose<<<GB, B, 0, stream>>>(bufA, bufB);   // bufB = dst^4
    k_compose<<<GB, B, 0, stream>>>(bufB, bufA);   // bufA = dst^8

    k_zero<<<(NN + 64 + 255) / 256, B, 0, stream>>>(counts, NN + 64);

    int* li = (int*)srow;  // srow dead after pooling
    k_count<<<GB, B, 0, stream>>>(fore, pin, bufA, li, counts);
    k_final<<<GB, B, 0, stream>>>(li, counts, out);
}